// PARTViT_70875550318749
// MI455X (gfx1250) — compile-verified
//
#include <hip/hip_runtime.h>
#include <hip/hip_bf16.h>

// Problem constants (from reference)
#define BB 64
#define NN 576
#define DD 384
#define KK 4096
#define NT 2
#define ROWS (BB * NN)        // 36864 patch rows total

typedef float v2f __attribute__((ext_vector_type(2)));
typedef float v8f __attribute__((ext_vector_type(8)));

// ---------------------------------------------------------------------------
// Kernel 1: per-patch projection  P[row][c] = dot(z[row], Wcol_c), c in 0..3
//   c=0: W[0, 0:D]   c=1: W[1, 0:D]   c=2: W[0, D:2D]   c=3: W[1, D:2D]
// One wave computes a 16-row tile via V_WMMA_F32_16X16X4_F32 accumulation.
// Block = 256 threads = 8 waves = 128 rows. Grid = ROWS/128 = 288.
// ---------------------------------------------------------------------------
__global__ __launch_bounds__(256) void partvit_project_wmma(
    const float* __restrict__ z,   // [ROWS, DD]
    const float* __restrict__ W,   // [NT, 2*DD] = [2, 768]
    float* __restrict__ P)         // [ROWS, 4]
{
    // LDS: B-matrix staging. Layout lds[(k/2)*32 + n*2 + (k&1)] = Wt[k][n]
    // where Wt[k][n] is the k-th element of projection column n (n<4), else 0.
    __shared__ float lds[(DD / 2) * 16 * 2];   // 6144 floats = 24 KB

    const int tid = threadIdx.x;
    for (int i = tid; i < DD * 16; i += 256) {
        const int p  = i & 1;
        const int n  = (i >> 1) & 15;
        const int kk = i >> 5;
        const int k  = kk * 2 + p;
        float v = 0.0f;
        if (n == 0)      v = W[k];              // W[0][k]
        else if (n == 1) v = W[2 * DD + k];     // W[1][k]
        else if (n == 2) v = W[DD + k];         // W[0][D+k]
        else if (n == 3) v = W[3 * DD + k];     // W[1][D+k]
        lds[i] = v;
    }
    __syncthreads();

    const int wave    = tid >> 5;               // 8 waves per block
    const int lane    = tid & 31;               // wave32
    const int half    = lane >> 4;              // 0: lanes 0-15, 1: lanes 16-31
    const int m       = lane & 15;              // M index within tile
    const int rowbase = blockIdx.x * 128 + wave * 16;

    const float* zrow = z + (size_t)(rowbase + m) * DD;

    v8f acc = {0.f, 0.f, 0.f, 0.f, 0.f, 0.f, 0.f, 0.f};

    // 96 WMMA steps of K=4 over D=384
    #pragma unroll 8
    for (int kc = 0; kc < DD; kc += 4) {
        // A (16x4 f32): VGPR0 = K=kc+2*half, VGPR1 = K=kc+2*half+1, M = lane%16
        v2f a = *(const v2f*)&zrow[kc + 2 * half];
        // B (4x16 f32): same K striping across halves, N = lane%16
        v2f b = *(const v2f*)&lds[(((kc >> 1) + half) * 16 + m) * 2];
        // D = A*B + C   (emits v_wmma_f32_16x16x4_f32)
        acc = __builtin_amdgcn_wmma_f32_16x16x4_f32(
            /*neg_a=*/false, a, /*neg_b=*/false, b,
            /*c_mod=*/(short)0, acc, /*reuse_a=*/false, /*reuse_b=*/false);
    }

    // C/D layout: VGPR r, lane -> M = r + half*8, N = lane%16.
    // Keep only the 4 live columns, store compactly: P[row][0..3].
    if (m < 4) {
        const int rowb = rowbase + half * 8;
        #pragma unroll
        for (int r = 0; r < 8; ++r) {
            P[(size_t)(rowb + r) * 4 + m] = acc[r];
        }
    }
}

// ---------------------------------------------------------------------------
// Kernel 2: per-pair gather + add + bias.
//   out[b,k,t] = P[b*N+i0][t] + P[b*N+i1][t+2] + bias[t]
// P is 576 KB -> fully L2-resident; all loads are 8-byte.
// ---------------------------------------------------------------------------
__global__ __launch_bounds__(256) void partvit_pair_gather(
    const int* __restrict__ indices,   // [BB, KK, 2] (int32)
    const float* __restrict__ P,       // [ROWS, 4]
    const float* __restrict__ bias,    // [NT]
    float* __restrict__ out)           // [BB, KK, NT]
{
    const int t = blockIdx.x * 256 + threadIdx.x;   // [0, BB*KK)
    const int b = t >> 12;                           // / KK
    const int i0 = indices[(size_t)t * 2 + 0];
    const int i1 = indices[(size_t)t * 2 + 1];
    const int row0 = b * NN + i0;
    const int row1 = b * NN + i1;

    const v2f p0 = *(const v2f*)&P[(size_t)row0 * 4];       // cols 0,1
    const v2f p1 = *(const v2f*)&P[(size_t)row1 * 4 + 2];   // cols 2,3

    v2f o;
    o.x = p0.x + p1.x + bias[0];
    o.y = p0.y + p1.y + bias[1];
    *(v2f*)&out[(size_t)t * 2] = o;
}

// ---------------------------------------------------------------------------
extern "C" void kernel_launch(void* const* d_in, const int* in_sizes, int n_in,
                              void* d_out, int out_size, void* d_ws, size_t ws_size,
                              hipStream_t stream) {
    const float* z       = (const float*)d_in[0];   // [B, N, D] f32
    const int*   indices = (const int*)d_in[1];     // [B, K, 2] int32
    const float* W       = (const float*)d_in[2];   // [2, 768] f32
    const float* bias    = (const float*)d_in[3];   // [2] f32

    float* P   = (float*)d_ws;                      // [ROWS, 4] = 576 KB scratch
    float* out = (float*)d_out;                     // [B, K, 2] f32

    // Projection: 288 blocks x 256 threads (8 waves x 16 rows = 128 rows/block)
    partvit_project_wmma<<<ROWS / 128, 256, 0, stream>>>(z, W, P);

    // Pair gather: one thread per (b, k)
    partvit_pair_gather<<<(BB * KK) / 256, 256, 0, stream>>>(indices, P, bias, out);
}